// QHadamard_24524263260380
// MI455X (gfx1250) — compile-verified
//
#include <hip/hip_runtime.h>

// FWHT-4096 for MI455X (gfx1250, wave32).
// One block per row (8192 rows). 4096 = 16^3 -> three H16 contractions,
// each done as 16 tiles of 16x16x16 matmul = 4 chained V_WMMA_F32_16X16X4_F32.
// fp32 WMMA => bit-exact vs the fp32 reference (H entries are +-1).
// Memory-bound: 256 MB total traffic -> ~11 us floor at 23.3 TB/s; single
// global read + single global write per element, everything else in LDS.

typedef float v2f __attribute__((ext_vector_type(2)));
typedef float v8f __attribute__((ext_vector_type(8)));
typedef float v4f __attribute__((ext_vector_type(4)));

#define FWHT_DIM 4096

__global__ __launch_bounds__(256) void fwht4096_wmma_kernel(
    const float* __restrict__ x, float* __restrict__ y, int nrows) {
  __shared__ float S0[FWHT_DIM];
  __shared__ float S1[FWHT_DIM];

  const int tid  = threadIdx.x;
  const int lane = tid & 31;
  const int wave = tid >> 5;
  const size_t row = blockIdx.x;
  if ((int)row >= nrows) return;  // uniform per block, no WMMA divergence

  const float* xr = x + row * (size_t)FWHT_DIM;
  float*       yr = y + row * (size_t)FWHT_DIM;

  // ---- Coalesced row load: 4096 floats, 256 threads * 4 x float4 ----
  {
    const v4f* xv = (const v4f*)xr;
    v4f*       sv = (v4f*)S0;
#pragma unroll
    for (int v = 0; v < 4; ++v) {
      int i = v * 256 + tid;
      sv[i] = xv[i];
    }
  }

  // ---- Per-lane H16 fragments, K sliced by 4 for 16x16x4 WMMA ----
  // Assumed f32 B(4x16) layout (mirror of documented A 16x4):
  //   vgpr0: lane<16 -> (K=4s+0, N=lane);  lane>=16 -> (K=4s+2, N=lane-16)
  //   vgpr1: lane<16 -> (K=4s+1, N=lane);  lane>=16 -> (K=4s+3, N=lane-16)
  // H is symmetric, so the same values serve as the A=H fragment
  // (documented A layout: vgpr0 = K{0|2}, vgpr1 = K{1|3}, M = lane%16).
  const int nh    = (lane < 16) ? lane : (lane - 16);  // N (or M) index
  const int kbase = (lane < 16) ? 0 : 2;               // K offset per lane half
  const int dM    = (lane < 16) ? 0 : 8;               // D-row offset per half
  v2f hfrag[4];
#pragma unroll
  for (int s = 0; s < 4; ++s) {
    int k0 = 4 * s + kbase;
    hfrag[s].x = (__builtin_popcount(k0 & nh) & 1) ? -1.0f : 1.0f;
    hfrag[s].y = (__builtin_popcount((k0 + 1) & nh) & 1) ? -1.0f : 1.0f;
  }

  __syncthreads();

  // ---- Stage 1: contract i0.  T1[t][m][n] = sum_k X[t][m][k] * H[k][n]
  // A = contiguous 16x16 tile of X (row stride 16), B = H fragment.
#pragma unroll
  for (int tt = 0; tt < 2; ++tt) {
    const int t = wave * 2 + tt;
    v8f c = {};
#pragma unroll
    for (int s = 0; s < 4; ++s) {
      v2f a;
      a.x = S0[t * 256 + nh * 16 + (4 * s + kbase + 0)];
      a.y = S0[t * 256 + nh * 16 + (4 * s + kbase + 1)];
      c = __builtin_amdgcn_wmma_f32_16x16x4_f32(false, a, false, hfrag[s],
                                                (short)0, c, false, false);
    }
#pragma unroll
    for (int r = 0; r < 8; ++r)
      S1[t * 256 + (r + dM) * 16 + nh] = c[r];
  }
  __syncthreads();

  // ---- Stage 2: contract i1.  T2[t][m][n] = sum_k H[m][k] * T1[t][k][n]
#pragma unroll
  for (int tt = 0; tt < 2; ++tt) {
    const int t = wave * 2 + tt;
    v8f c = {};
#pragma unroll
    for (int s = 0; s < 4; ++s) {
      v2f b;
      b.x = S1[t * 256 + (4 * s + kbase + 0) * 16 + nh];
      b.y = S1[t * 256 + (4 * s + kbase + 1) * 16 + nh];
      c = __builtin_amdgcn_wmma_f32_16x16x4_f32(false, hfrag[s], false, b,
                                                (short)0, c, false, false);
    }
#pragma unroll
    for (int r = 0; r < 8; ++r)
      S0[t * 256 + (r + dM) * 16 + nh] = c[r];
  }
  __syncthreads();

  // ---- Stage 3: contract i2.  Y[m][k1][n] = sum_k H[m][k] * T2[k][k1][n]
  // B rows stride 256 floats in LDS; D written straight to global with the
  // 1/64 scale fused. Each D-vgpr store covers two contiguous 64B segments.
#pragma unroll
  for (int tt = 0; tt < 2; ++tt) {
    const int k1 = wave * 2 + tt;
    v8f c = {};
#pragma unroll
    for (int s = 0; s < 4; ++s) {
      v2f b;
      b.x = S0[(4 * s + kbase + 0) * 256 + k1 * 16 + nh];
      b.y = S0[(4 * s + kbase + 1) * 256 + k1 * 16 + nh];
      c = __builtin_amdgcn_wmma_f32_16x16x4_f32(false, hfrag[s], false, b,
                                                (short)0, c, false, false);
    }
    const float scale = 1.0f / 64.0f;  // 1/sqrt(4096), X_H_SCALE = 1
#pragma unroll
    for (int r = 0; r < 8; ++r)
      yr[(r + dM) * 256 + k1 * 16 + nh] = c[r] * scale;
  }
}

extern "C" void kernel_launch(void* const* d_in, const int* in_sizes, int n_in,
                              void* d_out, int out_size, void* d_ws, size_t ws_size,
                              hipStream_t stream) {
  (void)n_in; (void)d_ws; (void)ws_size; (void)in_sizes;
  const float* x = (const float*)d_in[0];
  float* y = (float*)d_out;
  const int nrows = out_size / FWHT_DIM;  // 4 * 2048 = 8192
  fwht4096_wmma_kernel<<<nrows, 256, 0, stream>>>(x, y, nrows);
}